// Net_88218628260670
// MI455X (gfx1250) — compile-verified
//
#include <hip/hip_runtime.h>
#include <hip/hip_bf16.h>

typedef __attribute__((ext_vector_type(2))) float v2f;
typedef __attribute__((ext_vector_type(8))) float v8f;

// ---------------------------------------------------------------------------
// Degree / normalization kernels
// ---------------------------------------------------------------------------
__global__ void k_fill1(float* __restrict__ p, int n) {
    int i = blockIdx.x * blockDim.x + threadIdx.x;
    if (i < n) p[i] = 1.0f;                       // self-loop contributes 1 to degree
}

__global__ void k_deg(const int* __restrict__ dst, float* __restrict__ deg, int E) {
    int e = blockIdx.x * blockDim.x + threadIdx.x;
    if (e < E) atomicAdd(&deg[dst[e]], 1.0f);
}

__global__ void k_rsqrt(float* __restrict__ d, int n) {
    int i = blockIdx.x * blockDim.x + threadIdx.x;
    if (i < n) {
        float v = d[i];
        d[i] = (v > 0.0f) ? rsqrtf(v) : 0.0f;
    }
}

// ---------------------------------------------------------------------------
// conv1 linear (K=5, scalar) fused with self-loop agg init:
//   lin[n,f] = sum_k x[n,k]*W1[k,f];  agg[n,f] = dinv[n]^2 * lin[n,f]
// ---------------------------------------------------------------------------
__global__ void k_lin1(const float* __restrict__ x, const float* __restrict__ W1,
                       const float* __restrict__ dinv,
                       float* __restrict__ lin, float* __restrict__ agg, int N) {
    int idx = blockIdx.x * blockDim.x + threadIdx.x;
    if (idx >= N * 64) return;
    int n = idx >> 6, f = idx & 63;
    float acc = 0.0f;
#pragma unroll
    for (int k = 0; k < 5; ++k) acc += x[n * 5 + k] * W1[k * 64 + f];
    lin[idx] = acc;
    float di = dinv[n];
    agg[idx] = di * di * acc;
}

// agg[n,f] = dinv[n]^2 * h[n,f]   (self-loop term, also initializes agg)
__global__ void k_selfinit(const float* __restrict__ h, const float* __restrict__ dinv,
                           float* __restrict__ agg, int N) {
    int idx = blockIdx.x * blockDim.x + threadIdx.x;
    if (idx >= N * 64) return;
    int n = idx >> 6;
    float di = dinv[n];
    agg[idx] = di * di * h[idx];
}

// Edge scatter: agg[dst] += dinv[src]*dinv[dst]*h[src].  16 threads / edge,
// float4 gather per thread; atomics resolve in L2 (node table fits in 192MB L2).
__global__ void k_edgeagg(const int* __restrict__ src, const int* __restrict__ dst,
                          const float* __restrict__ dinv, const float* __restrict__ h,
                          float* __restrict__ agg, int E) {
    int t = blockIdx.x * blockDim.x + threadIdx.x;
    if (t >= E * 16) return;
    int e = t >> 4;
    int q = t & 15;
    int s = src[e], d = dst[e];
    float w = dinv[s] * dinv[d];
    const float4 v = *(const float4*)(h + (size_t)s * 64 + q * 4);
    float* base = agg + (size_t)d * 64 + q * 4;
    atomicAdd(base + 0, w * v.x);
    atomicAdd(base + 1, w * v.y);
    atomicAdd(base + 2, w * v.z);
    atomicAdd(base + 3, w * v.w);
}

// ---------------------------------------------------------------------------
// WMMA fp32 GEMM: out[N,64] = act(act_in(in) @ W[64,64] (+ bias))
// PREB: apply relu(a + preb[k]) to the A operand on load (fuses the previous
// layer's bias+ReLU; VALU co-executes with the XDL WMMAs).
// Block = 128 threads (4 waves); wave w owns 16-column tile w; 16 rows/block.
// ---------------------------------------------------------------------------
template <bool PREB, bool BIAS, bool RELU>
__global__ void k_gemm64(const float* __restrict__ A, const float* __restrict__ preb,
                         const float* __restrict__ W, const float* __restrict__ bias,
                         float* __restrict__ out, int N) {
    const int lane = threadIdx.x & 31;
    const int wave = threadIdx.x >> 5;       // 0..3 = column tile
    const int m    = lane & 15;
    const int half = lane >> 4;              // 0 or 1 (K sub-pair)
    const int n    = wave * 16 + m;          // global output column
    const int row0 = blockIdx.x * 16;

    int ra = row0 + m;
    if (ra >= N) ra = N - 1;                 // clamp (no EXEC divergence at WMMA)
    const float* arow = A + (size_t)ra * 64;

    v8f c = {};
#pragma unroll
    for (int kb = 0; kb < 16; ++kb) {
        const int k0 = kb * 4 + half * 2;
        v2f a, b;
        a.x = arow[k0];
        a.y = arow[k0 + 1];
        if (PREB) {
            a.x += preb[k0];
            a.y += preb[k0 + 1];
            a.x = a.x > 0.0f ? a.x : 0.0f;
            a.y = a.y > 0.0f ? a.y : 0.0f;
        }
        b.x = W[(size_t)k0 * 64 + n];
        b.y = W[(size_t)(k0 + 1) * 64 + n];
        c = __builtin_amdgcn_wmma_f32_16x16x4_f32(false, a, false, b, (short)0, c,
                                                  false, false);
    }
    const float bv = BIAS ? bias[n] : 0.0f;
#pragma unroll
    for (int j = 0; j < 8; ++j) {
        int row = row0 + j + half * 8;
        if (row < N) {
            float v = c[j] + bv;
            if (RELU) v = v > 0.0f ? v : 0.0f;
            out[(size_t)row * 64 + n] = v;
        }
    }
}

// fc1: out = relu(concat(relu(H + preb), G) @ Wf1[67,64] + bf1)
// WMMA over K=0..63 (with conv2's bias+ReLU fused into the A-load),
// scalar tail for the 3 global-feature K rows.
__global__ void k_fc1(const float* __restrict__ H, const float* __restrict__ preb,
                      const float* __restrict__ G,
                      const float* __restrict__ Wf1, const float* __restrict__ bf1,
                      float* __restrict__ out, int N) {
    const int lane = threadIdx.x & 31;
    const int wave = threadIdx.x >> 5;
    const int m    = lane & 15;
    const int half = lane >> 4;
    const int n    = wave * 16 + m;
    const int row0 = blockIdx.x * 16;

    int ra = row0 + m;
    if (ra >= N) ra = N - 1;
    const float* arow = H + (size_t)ra * 64;

    v8f c = {};
#pragma unroll
    for (int kb = 0; kb < 16; ++kb) {
        const int k0 = kb * 4 + half * 2;
        v2f a, b;
        a.x = arow[k0] + preb[k0];
        a.y = arow[k0 + 1] + preb[k0 + 1];
        a.x = a.x > 0.0f ? a.x : 0.0f;
        a.y = a.y > 0.0f ? a.y : 0.0f;
        b.x = Wf1[(size_t)k0 * 64 + n];
        b.y = Wf1[(size_t)(k0 + 1) * 64 + n];
        c = __builtin_amdgcn_wmma_f32_16x16x4_f32(false, a, false, b, (short)0, c,
                                                  false, false);
    }
    const float bv = bf1[n];
    const float wg0 = Wf1[(size_t)64 * 64 + n];
    const float wg1 = Wf1[(size_t)65 * 64 + n];
    const float wg2 = Wf1[(size_t)66 * 64 + n];
#pragma unroll
    for (int j = 0; j < 8; ++j) {
        int row = row0 + j + half * 8;
        if (row < N) {
            const float* g = G + (size_t)row * 3;
            float v = c[j] + bv + g[0] * wg0 + g[1] * wg1 + g[2] * wg2;
            v = v > 0.0f ? v : 0.0f;
            out[(size_t)row * 64 + n] = v;
        }
    }
}

// Output layer: out[N,30] = H[N,64] @ Wo[64,30] + bo.  Block = 64 threads
// (2 waves -> columns 0..31); columns >= 30 masked at B-load and store.
__global__ void k_out(const float* __restrict__ H, const float* __restrict__ Wo,
                      const float* __restrict__ bo, float* __restrict__ out, int N) {
    const int lane = threadIdx.x & 31;
    const int wave = threadIdx.x >> 5;       // 0..1
    const int m    = lane & 15;
    const int half = lane >> 4;
    const int n    = wave * 16 + m;          // 0..31; valid if < 30
    const int row0 = blockIdx.x * 16;
    const bool nv  = (n < 30);

    int ra = row0 + m;
    if (ra >= N) ra = N - 1;
    const float* arow = H + (size_t)ra * 64;

    v8f c = {};
#pragma unroll
    for (int kb = 0; kb < 16; ++kb) {
        const int k0 = kb * 4 + half * 2;
        v2f a, b;
        a.x = arow[k0];
        a.y = arow[k0 + 1];
        b.x = nv ? Wo[(size_t)k0 * 30 + n] : 0.0f;
        b.y = nv ? Wo[(size_t)(k0 + 1) * 30 + n] : 0.0f;
        c = __builtin_amdgcn_wmma_f32_16x16x4_f32(false, a, false, b, (short)0, c,
                                                  false, false);
    }
    if (nv) {
        const float bv = bo[n];
#pragma unroll
        for (int j = 0; j < 8; ++j) {
            int row = row0 + j + half * 8;
            if (row < N) out[(size_t)row * 30 + n] = c[j] + bv;
        }
    }
}

// ---------------------------------------------------------------------------
// Launch
// ---------------------------------------------------------------------------
extern "C" void kernel_launch(void* const* d_in, const int* in_sizes, int n_in,
                              void* d_out, int out_size, void* d_ws, size_t ws_size,
                              hipStream_t stream) {
    const float* x     = (const float*)d_in[0];
    const int*   ei    = (const int*)  d_in[1];
    const float* globf = (const float*)d_in[2];
    const float* W1    = (const float*)d_in[3];
    const float* b1    = (const float*)d_in[4];
    const float* W2    = (const float*)d_in[5];
    const float* b2    = (const float*)d_in[6];
    const float* Wf1   = (const float*)d_in[7];
    const float* bf1   = (const float*)d_in[8];
    const float* Wf2   = (const float*)d_in[9];
    const float* bf2   = (const float*)d_in[10];
    const float* Wo    = (const float*)d_in[11];
    const float* bo    = (const float*)d_in[12];
    float* out = (float*)d_out;

    const int N = in_sizes[0] / 5;
    const int E = in_sizes[1] / 2;
    const int* src = ei;
    const int* dst = ei + E;

    // Workspace layout: dinv[N] | bufA[N*64] | bufB[N*64]  (~51.6 MB)
    float* w = (float*)d_ws;
    float* dinv = w;
    size_t off = ((size_t)N + 255) & ~(size_t)255;
    float* bufA = w + off;
    float* bufB = bufA + (size_t)N * 64;

    const int T = 256;
    const int gN   = (N + T - 1) / T;
    const int gE   = (E + T - 1) / T;
    const int gNF  = (N * 64 + T - 1) / T;
    const int gE16 = (E * 16 + T - 1) / T;
    const int ntiles = (N + 15) / 16;

    // symmetric GCN normalization: deg = 1 + indeg; dinv = rsqrt(deg)
    k_fill1<<<gN, T, 0, stream>>>(dinv, N);
    k_deg  <<<gE, T, 0, stream>>>(dst, dinv, E);
    k_rsqrt<<<gN, T, 0, stream>>>(dinv, N);

    // conv1: lin (+fused self-loop init) -> edge scatter (bias+relu deferred)
    k_lin1   <<<gNF,  T, 0, stream>>>(x, W1, dinv, bufA, bufB, N);
    k_edgeagg<<<gE16, T, 0, stream>>>(src, dst, dinv, bufA, bufB, E);

    // conv2: WMMA GEMM with conv1 bias+ReLU fused into A-load -> aggregation
    k_gemm64<true, false, false><<<ntiles, 128, 0, stream>>>(bufB, b1, W2, nullptr,
                                                             bufA, N);
    k_selfinit<<<gNF,  T, 0, stream>>>(bufA, dinv, bufB, N);
    k_edgeagg <<<gE16, T, 0, stream>>>(src, dst, dinv, bufA, bufB, E);

    // MLP head: fc1 fuses conv2 bias+ReLU into A-load
    k_fc1<<<ntiles, 128, 0, stream>>>(bufB, b2, globf, Wf1, bf1, bufA, N);
    k_gemm64<false, true, true><<<ntiles, 128, 0, stream>>>(bufA, nullptr, Wf2, bf2,
                                                            bufB, N);
    k_out<<<ntiles, 64, 0, stream>>>(bufB, Wo, bo, out, N);
}